// PCENLayer_79929341378573
// MI455X (gfx1250) — compile-verified
//
#include <hip/hip_runtime.h>
#include <hip/hip_bf16.h>

// PCEN (32, 8192, 64) f32 — chunked-scan EMA + WMMA triangular matmul.
// Phase 1: per-(b,c,chunk16) carry (1,048,576 threads, coalesced).
// Phase 2: per-(b,c) affine scan over 512 chunk carries -> a_in per chunk.
// Phase 3: per-wave 16x16 tile: ema = W @ X + d (x) a_in via
//          V_WMMA_F32_16X16X4_F32, then power-law output.

constexpr int kB   = 32;
constexpr int kT   = 8192;
constexpr int kC   = 64;
constexpr int kL   = 16;            // chunk length (WMMA M)
constexpr int kNCH = kT / kL;       // 512 chunks per sequence
constexpr int kPad = 15;            // zero-padding for triangular mask
constexpr float kFloor = 1e-06f;

typedef float v2f __attribute__((ext_vector_type(2)));
typedef float v8f __attribute__((ext_vector_type(8)));

__device__ __forceinline__ float clamp01f(float v) {
    return fminf(fmaxf(v, 0.0f), 1.0f);
}
// Native gfx transcendentals: v_log_f32 (log2), v_exp_f32 (exp2).
__device__ __forceinline__ float flog2(float v) { return __builtin_amdgcn_logf(v); }
__device__ __forceinline__ float fexp2(float v) { return __builtin_amdgcn_exp2f(v); }

// ---------------------------------------------------------------- Phase 1 --
// carry[(b*kNCH + j)*kC + c] = sum_{i=0..15} k * q^{15-i} * x[b][16j+i][c]
__global__ void __launch_bounds__(256)
pcen_chunk_carry(const float* __restrict__ x,
                 const float* __restrict__ smooth,
                 float* __restrict__ carry)
{
    const int gt = blockIdx.x * blockDim.x + threadIdx.x;   // kB*kNCH*kC threads
    const float k = clamp01f(smooth[0]);
    const float q = 1.0f - k;

    const int c  = gt & (kC - 1);
    const int bj = gt >> 6;                 // b*kNCH + j
    const int j  = bj & (kNCH - 1);
    const int b  = bj >> 9;                 // kNCH == 512

    const float* xp = x + (size_t)(b * kT + j * kL) * kC + c;
    __builtin_prefetch(xp + 8 * kC, 0, 0);  // global_prefetch_b8: 2nd half of chunk

    float a = 0.0f;
#pragma unroll
    for (int i = 0; i < kL; ++i)
        a = fmaf(q, a, k * xp[(size_t)i * kC]);   // a = q*a + k*x
    carry[gt] = a;
}

// ---------------------------------------------------------------- Phase 2 --
// a_in(0) = x[b][0][c];  a_in(j+1) = q^16 * a_in(j) + carry(j)
__global__ void __launch_bounds__(256)
pcen_carry_scan(const float* __restrict__ x,
                const float* __restrict__ smooth,
                const float* __restrict__ carry,
                float* __restrict__ ain)
{
    const int gt = blockIdx.x * blockDim.x + threadIdx.x;   // kB*kC threads
    const float k = clamp01f(smooth[0]);
    const float q = 1.0f - k;
    const float q2 = q * q, q4 = q2 * q2, q8 = q4 * q4, q16 = q8 * q8;

    const int c = gt & (kC - 1);
    const int b = gt >> 6;

    float a = x[(size_t)b * kT * kC + c];   // init state a_{-1} = x[:,0,:]
    const float* cp = carry + (size_t)b * kNCH * kC + c;
    float*       ap = ain   + (size_t)b * kNCH * kC + c;

    for (int j = 0; j < kNCH; ++j) {
        ap[(size_t)j * kC] = a;
        a = fmaf(q16, a, cp[(size_t)j * kC]);
    }
}

// ---------------------------------------------------------------- Phase 3 --
// One wave per (b, chunk, 16-channel tile). ema tile via 4x WMMA_F32_16X16X4_F32.
// Triangular mask is data-driven: pwPad[t] = 0 for t<15, k*q^{t-15} for t>=15,
// so A[i][j] = pwPad[i-j+15] needs no predication (EXEC stays all-1s).
__global__ void __launch_bounds__(256)
pcen_wmma_out(const float* __restrict__ x,
              const float* __restrict__ smooth,
              const float* __restrict__ alpha,
              const float* __restrict__ delta,
              const float* __restrict__ root,
              const float* __restrict__ ain,
              float* __restrict__ out)
{
    __shared__ float pwPad[kPad + kL + 1];   // 32 entries

    const int tid = threadIdx.x;
    const float k = clamp01f(smooth[0]);
    const float q = 1.0f - k;

    if (tid < kPad + kL + 1) {
        float p = 0.0f;
        if (tid >= kPad) {
            p = k;
            for (int t = kPad; t < tid; ++t) p *= q;   // k * q^{tid-kPad}
        }
        pwPad[tid] = p;
    }
    __syncthreads();

    const int wave = tid >> 5;              // 8 waves / block
    const int lane = tid & 31;
    const int hi   = lane >> 4;             // lane half (K / row-half select)
    const int li   = lane & 15;             // A row index == C/D column index

    const int tileId = blockIdx.x * 8 + wave;        // [0, kB*kNCH*4)
    const int ntile  = tileId & 3;                   // 16-channel tile
    const int bj     = tileId >> 2;                  // b*kNCH + j
    const int j      = bj & (kNCH - 1);
    const int b      = bj >> 9;

    const int    c    = ntile * 16 + li;             // channel
    const int    t0   = j * kL;                      // chunk start time
    const size_t base = (size_t)(b * kT + t0) * kC + c;

    // per-channel parameters
    const float al  = fminf(alpha[c], 1.0f);
    const float de  = delta[c];
    const float ir  = 1.0f / fmaxf(root[c], 1.0f);
    const float dpw = fexp2(ir * flog2(de));         // delta^(1/root)

    const float a_in = ain[(size_t)(b * kNCH + j) * kC + c];

    // C/D layout: VGPR r, lanes 0-15 -> row r, lanes 16-31 -> row r+8.
    // Seed accumulator with decay term: C[i][c] = q^{i+1} * a_in.
    const float q2 = q * q, q4 = q2 * q2, q8 = q4 * q4;
    float dr = q * (hi ? q8 : 1.0f);                 // q^{8*hi + 1}
    v8f acc;
#pragma unroll
    for (int r = 0; r < 8; ++r) { acc[r] = dr * a_in; dr *= q; }

    // Accumulate ema = W @ X over K in steps of 4.
    const int ebase = li - 2 * hi + kPad;            // padded index at kb=0
#pragma unroll
    for (int kb = 0; kb < 4; ++kb) {
        // A 16x4 f32 layout: lane li = row; VGPR0 holds K = 2*hi, VGPR1 K = 2*hi+1.
        const int e = ebase - 4 * kb;
        v2f A = {pwPad[e], pwPad[e - 1]};            // unconditional ds_load_b32

        // B 4x16 f32: row K striped across lanes; VGPR0 K=2*hi, VGPR1 K=2*hi+1.
        v2f Bv;
        Bv[0] = x[base + (size_t)(4 * kb + 2 * hi) * kC];
        Bv[1] = x[base + (size_t)(4 * kb + 2 * hi + 1) * kC];

        acc = __builtin_amdgcn_wmma_f32_16x16x4_f32(
            /*neg_a=*/false, A, /*neg_b=*/false, Bv,
            /*c_mod=*/(short)0, acc, /*reuse_a=*/false, /*reuse_b=*/false);
    }

    // Elementwise PCEN output from the ema tile.
#pragma unroll
    for (int r = 0; r < 8; ++r) {
        const size_t idx = base + (size_t)(r + 8 * hi) * kC;
        const float xv = x[idx];                              // L2-resident re-read
        const float ep = fexp2(al * flog2(kFloor + acc[r]));
        const float y  = xv / ep + de;
        out[idx] = fexp2(ir * flog2(y)) - dpw;
    }
}

// ---------------------------------------------------------------- launch ----
extern "C" void kernel_launch(void* const* d_in, const int* in_sizes, int n_in,
                              void* d_out, int out_size, void* d_ws, size_t ws_size,
                              hipStream_t stream)
{
    const float* x      = (const float*)d_in[0];   // (32, 8192, 64) f32
    const float* smooth = (const float*)d_in[1];   // (1,)
    const float* alpha  = (const float*)d_in[2];   // (64,)
    const float* delta  = (const float*)d_in[3];   // (64,)
    const float* root   = (const float*)d_in[4];   // (64,)
    float* out = (float*)d_out;

    float* carry = (float*)d_ws;                               // 4 MB
    float* ain   = carry + (size_t)kB * kNCH * kC;             // 4 MB

    pcen_chunk_carry<<<(kB * kNCH * kC) / 256, 256, 0, stream>>>(x, smooth, carry);
    pcen_carry_scan <<<(kB * kC) / 256,        256, 0, stream>>>(x, smooth, carry, ain);
    pcen_wmma_out   <<<(kB * kNCH * 4) / 8,    256, 0, stream>>>(x, smooth, alpha, delta,
                                                                 root, ain, out);
}